// GeometricBilinearLayer_61564061221557
// MI455X (gfx1250) — compile-verified
//
#include <hip/hip_runtime.h>
#include <cstddef>

typedef __attribute__((ext_vector_type(2))) float v2f;
typedef __attribute__((ext_vector_type(8))) float v8f;

// ================= compile-time G(3,0,1) Cayley tables =================
// Blades as 4-bit masks (bit p == e_p present). Metric: e0^2=0, e1..3^2=+1.
namespace ga {
constexpr int pc(int x){ int n=0; while(x){ n += x & 1; x >>= 1; } return n; }
// sign of bringing e_A e_B into canonical order
constexpr int rsign(int a, int b){ int s=0; a >>= 1; while(a){ s += pc(a & b); a >>= 1; } return (s & 1) ? -1 : 1; }
struct Tables {
  int maskof[16]; int idxof[16];
  float gp[16][16]; int gpk[16][16];   // geometric product: 192 nonzeros
  float jn[16][16]; int jnk[16][16];   // PGA join:           81 nonzeros
  constexpr Tables()
    : maskof{0, 1,2,4,8, 3,5,9,6,10,12, 7,11,13,14, 15},
      idxof{}, gp{}, gpk{}, jn{}, jnk{} {
    for (int i=0;i<16;i++) idxof[maskof[i]] = i;
    for (int i=0;i<16;i++) for (int j=0;j<16;j++) {
      const int a = maskof[i], b = maskof[j];
      if (!(a & b & 1)) {                       // shared e0 -> 0 (degenerate)
        gp[i][j]  = (float)rsign(a,b);
        gpk[i][j] = idxof[a ^ b];
      }
      const int ca = 15 & ~a, cb = 15 & ~b;     // right complements (duals)
      if ((ca & cb) == 0) {                     // dual(x)^dual(y) != 0
        const int si = rsign(a,ca), sj = rsign(b,cb);
        const int m  = ca | cb;  const int sw = rsign(ca,cb);
        const int km = 15 & ~m;  const int su = rsign(km,m); // undual = D^T
        jn[i][j]  = (float)(si*sj*sw*su);
        jnk[i][j] = idxof[km];
      }
    }
  }
};
constexpr Tables T{};
}
__constant__ int GRADE16[16] = {0,1,1,1,1,2,2,2,2,2,2,3,3,3,3,4};

// ================= LDS layout (floats), one 16-token tile =================
#define TSTR 68                       // padded channel stride (bank-spread)
#define HSTR 36                       // padded h stride for px/py
#define XS_OFF   0                    // xs[blade][t][c]   : 16*16*68
#define PA_OFF   (16*16*TSTR)         // pX[blade][t][h]   : 16*16*36
#define PB_OFF   (PA_OFF + 16*16*HSTR)
#define FS_OFF   (PB_OFF + 16*16*HSTR)// feats[blade][t][c]: 16*16*68
#define LDS_FLOATS (FS_OFF + 16*16*TSTR)   // 53248 floats = 208 KB < 320 KB/WGP

__global__ __launch_bounds__(256, 1)
void gatr_bilinear_kernel(const float* __restrict__ x, const float* __restrict__ ref,
                          const float* __restrict__ wPX, const float* __restrict__ wPY,
                          const float* __restrict__ wJX, const float* __restrict__ wJY,
                          const float* __restrict__ wF, float* __restrict__ out)
{
  extern __shared__ float lds[];
  float* xs = lds + XS_OFF;
  float* pA = lds + PA_OFF;
  float* pB = lds + PB_OFF;
  float* fs = lds + FS_OFF;

  const int tid  = threadIdx.x;
  const int tok0 = blockIdx.x * 16;

  // ---- stage x tile: coalesced float4 reads (512B/wave), transpose to xs[k][t][c]
  for (int q = tid; q < 4096; q += 256) {
    const int t  = q >> 8;
    const int c  = (q >> 2) & 63;
    const int k4 = q & 3;
    const float4 v = *(const float4*)(x + (((size_t)(tok0 + t))*64 + c)*16 + k4*4);
    xs[((k4*4+0)*16 + t)*TSTR + c] = v.x;
    xs[((k4*4+1)*16 + t)*TSTR + c] = v.y;
    xs[((k4*4+2)*16 + t)*TSTR + c] = v.z;
    xs[((k4*4+3)*16 + t)*TSTR + c] = v.w;
  }
  __syncthreads();

  const int wave = tid >> 5, lane = tid & 31;
  const int lh = lane & 15, lu = lane >> 4;   // A/B frag: lane=row, K split at lane 16

  for (int branch = 0; branch < 2; ++branch) {
    const float* WX = branch ? wJX : wPX;
    const float* WY = branch ? wJY : wPY;

    // ---- equilin X & Y: 2 mats x 16 blades x 2 h-tiles = 64 tiles / 8 waves
    for (int it = 0; it < 8; ++it) {
      const int tile  = wave + 8*it;
      const int mat   = tile >> 5;
      const int blade = (tile >> 1) & 15;
      const int ht    = tile & 1;
      const float* W  = mat ? WY : WX;
      const int g     = GRADE16[blade];
      v8f acc = {};
      const float* arow = xs + (blade*16 + lh)*TSTR + 2*lu;                  // A(M=t,K=c)
      const float* brow = W + ((size_t)g*32 + ht*16 + lh)*64 + 2*lu;         // B(K=c,N=h)=w[g][h][c]
#pragma unroll
      for (int kk = 0; kk < 16; ++kk) {                                      // K = 64 = 16 x 4
        const v2f a = *(const v2f*)(arow + 4*kk);
        const v2f b = *(const v2f*)(brow + 4*kk);
        acc = __builtin_amdgcn_wmma_f32_16x16x4_f32(false, a, false, b, (short)0, acc, false, false);
      }
      float* dst = (mat ? pB : pA) + blade*16*HSTR + ht*16 + lh;
#pragma unroll
      for (int v = 0; v < 8; ++v) dst[(v + 8*lu)*HSTR] = acc[v];             // D: M = v + 8*(lane>=16)
    }
    __syncthreads();

    // ---- sparse bilinear per (t,h): 512 pairs, 2 per thread, tables unrolled to +-FMAs
    for (int pp = 0; pp < 2; ++pp) {
      const int p = tid + 256*pp;
      const int t = p >> 5, h = p & 31;
      float a[16], b[16], r[16];
#pragma unroll
      for (int i = 0; i < 16; ++i) {
        a[i] = pA[(i*16 + t)*HSTR + h];
        b[i] = pB[(i*16 + t)*HSTR + h];
        r[i] = 0.f;
      }
      if (branch == 0) {
#pragma unroll
        for (int i = 0; i < 16; ++i)
#pragma unroll
          for (int j = 0; j < 16; ++j) {
            const float s = ga::T.gp[i][j];
            if (s != 0.f) r[ga::T.gpk[i][j]] += s * a[i] * b[j];
          }
      } else {
#pragma unroll
        for (int i = 0; i < 16; ++i)
#pragma unroll
          for (int j = 0; j < 16; ++j) {
            const float s = ga::T.jn[i][j];
            if (s != 0.f) r[ga::T.jnk[i][j]] += s * a[i] * b[j];
          }
      }
      const float scale = (branch == 0) ? 1.f : ref[((size_t)(tok0 + t))*16 + 15];
#pragma unroll
      for (int k = 0; k < 16; ++k)
        fs[(k*16 + t)*TSTR + branch*32 + h] = r[k] * scale;
    }
    __syncthreads();
  }

  // ---- final equilin: 16 blades x 4 o-tiles, K=64; D goes straight to HBM
  for (int it = 0; it < 8; ++it) {
    const int tile  = wave + 8*it;
    const int blade = tile >> 2;
    const int ot    = tile & 3;
    const int g     = GRADE16[blade];
    v8f acc = {};
    const float* arow = fs + (blade*16 + lh)*TSTR + 2*lu;
    const float* brow = wF + ((size_t)g*64 + ot*16 + lh)*64 + 2*lu;
#pragma unroll
    for (int kk = 0; kk < 16; ++kk) {
      const v2f a = *(const v2f*)(arow + 4*kk);
      const v2f b = *(const v2f*)(brow + 4*kk);
      acc = __builtin_amdgcn_wmma_f32_16x16x4_f32(false, a, false, b, (short)0, acc, false, false);
    }
#pragma unroll
    for (int v = 0; v < 8; ++v)
      out[(((size_t)(tok0 + v + 8*lu))*64 + ot*16 + lh)*16 + blade] = acc[v];
  }
}

extern "C" void kernel_launch(void* const* d_in, const int* in_sizes, int n_in,
                              void* d_out, int out_size, void* d_ws, size_t ws_size,
                              hipStream_t stream) {
  (void)n_in; (void)out_size; (void)d_ws; (void)ws_size;
  const float* x   = (const float*)d_in[0];
  const float* ref = (const float*)d_in[1];
  const float* wPX = (const float*)d_in[2];
  const float* wPY = (const float*)d_in[3];
  const float* wJX = (const float*)d_in[4];
  const float* wJY = (const float*)d_in[5];
  const float* wF  = (const float*)d_in[6];

  const int ntok = in_sizes[0] / (64 * 16);   // B*N = 32768
  const int grid = ntok / 16;                 // one 16-token tile per WG
  const size_t shmem = LDS_FLOATS * sizeof(float);
  gatr_bilinear_kernel<<<grid, 256, shmem, stream>>>(x, ref, wPX, wPY, wJX, wJY, wF, (float*)d_out);
}